// ChebyASPIRELayer_56229711839494
// MI455X (gfx1250) — compile-verified
//
#include <hip/hip_runtime.h>
#include <hip/hip_bf16.h>

typedef __attribute__((ext_vector_type(2))) float v2f;
typedef __attribute__((ext_vector_type(8))) float v8f;

#define D 64
#define ND_PER_ROW 64

// -------------------------------------------------------------------------
// out[n*64+d] = bias[d]
__global__ void cheb_init_bias(float* __restrict__ out, const float* __restrict__ bias,
                               int total) {
    int i = blockIdx.x * blockDim.x + threadIdx.x;
    if (i < total) out[i] = bias[i & 63];
}

// zero a float4-aligned buffer (count4 = number of float4 elements)
__global__ void cheb_zero4(float4* __restrict__ p, int count4) {
    int i = blockIdx.x * blockDim.x + threadIdx.x;
    if (i < count4) p[i] = make_float4(0.f, 0.f, 0.f, 0.f);
}

// dst = -src (float4 vectorized): folds the "- T_{k-2}" into the init
__global__ void cheb_negcopy4(float4* __restrict__ dst, const float4* __restrict__ src,
                              int count4) {
    int i = blockIdx.x * blockDim.x + threadIdx.x;
    if (i < count4) {
        float4 v = src[i];
        dst[i] = make_float4(-v.x, -v.y, -v.z, -v.w);
    }
}

// -------------------------------------------------------------------------
// Sparse propagate: outT[dst_e] += scale * w_e * h[src_e]
// 16 lanes per edge, each lane handles a float4 chunk of the 64-wide row.
// A 16-lane group writes 64 consecutive floats -> coalesced atomic bursts
// that stay resident in the 192MB L2.
__global__ void cheb_propagate(const float* __restrict__ h,
                               const int* __restrict__ edge_index,
                               const float* __restrict__ edge_weight,
                               float* __restrict__ outT,
                               float scale, int nEdges) {
    int t = blockIdx.x * blockDim.x + threadIdx.x;
    int e = t >> 4;
    if (e >= nEdges) return;
    int c = (t & 15) << 2;                  // float4 chunk start column
    int src = edge_index[e];                // edge_index[0][e]
    int dst = edge_index[nEdges + e];       // edge_index[1][e]
    float w = edge_weight[e] * scale;
    const float4 v = *reinterpret_cast<const float4*>(h + (size_t)src * ND_PER_ROW + c);
    float* o = outT + (size_t)dst * ND_PER_ROW + c;
    atomicAdd(o + 0, w * v.x);
    atomicAdd(o + 1, w * v.y);
    atomicAdd(o + 2, w * v.z);
    atomicAdd(o + 3, w * v.w);
}

// -------------------------------------------------------------------------
// out[rows, 64] += T[rows, 64] @ W[64, 64]   using V_WMMA_F32_16X16X4_F32.
// Block = 256 threads = 8 waves. Waves tile 32 rows x 64 cols:
//   wave>>2 selects the 16-row tile, wave&3 selects the 16-col tile.
// W (16KB) is staged in LDS once per block.
__global__ __launch_bounds__(256)
void cheb_gemm_acc(const float* __restrict__ T, const float* __restrict__ W,
                   float* __restrict__ out, int nRows) {
    __shared__ float sW[D * D];
    int tid = threadIdx.x;

    // cooperative stage of the 64x64 weight into LDS (float4 x 4 per thread)
    const float4* Wv = reinterpret_cast<const float4*>(W);
    float4* sWv = reinterpret_cast<float4*>(sW);
    #pragma unroll
    for (int i = tid; i < (D * D) / 4; i += 256) sWv[i] = Wv[i];
    __syncthreads();

    int wave = tid >> 5;
    int lane = tid & 31;
    int rowBase = (blockIdx.x * 2 + (wave >> 2)) * 16;
    if (rowBase >= nRows) return;           // wave-uniform: EXEC stays all-1s
    int colBase = (wave & 3) * 16;

    int lhalf = lane >> 4;                  // 0: lanes 0-15, 1: lanes 16-31
    int lmod  = lane & 15;
    const float* aRowPtr = T + (size_t)(rowBase + lmod) * ND_PER_ROW;

    v8f c = {0.f, 0.f, 0.f, 0.f, 0.f, 0.f, 0.f, 0.f};

    #pragma unroll
    for (int kb = 0; kb < D; kb += 4) {
        // A fragment (16x4): lane<16 -> K = kb,kb+1 ; lane>=16 -> K = kb+2,kb+3
        v2f a = *reinterpret_cast<const v2f*>(aRowPtr + kb + lhalf * 2);
        // B fragment (4x16): v0 -> rows kb (lanes 0-15) / kb+2 (lanes 16-31),
        //                    v1 -> rows kb+1 / kb+3
        v2f b;
        b.x = sW[(kb + lhalf * 2 + 0) * D + colBase + lmod];
        b.y = sW[(kb + lhalf * 2 + 1) * D + colBase + lmod];
        c = __builtin_amdgcn_wmma_f32_16x16x4_f32(false, a, false, b,
                                                  (short)0, c, false, false);
    }

    // C/D layout: VGPR r -> M = r + 8*lhalf, N = lane&15. Tile exclusively
    // owned by this wave, so plain read-modify-write accumulate.
    #pragma unroll
    for (int r = 0; r < 8; ++r) {
        size_t idx = (size_t)(rowBase + r + lhalf * 8) * ND_PER_ROW + colBase + lmod;
        out[idx] += c[r];
    }
}

// -------------------------------------------------------------------------
extern "C" void kernel_launch(void* const* d_in, const int* in_sizes, int n_in,
                              void* d_out, int out_size, void* d_ws, size_t ws_size,
                              hipStream_t stream) {
    const float* x     = (const float*)d_in[0];
    const int*   ei    = (const int*)d_in[1];
    const float* ew    = (const float*)d_in[2];
    const float* W     = (const float*)d_in[3];   // (K, 64, 64)
    const float* bias  = (const float*)d_in[4];
    float* out = (float*)d_out;

    const int nNodes = in_sizes[0] / D;
    const int nEdges = in_sizes[2];
    const int K      = in_sizes[3] / (D * D);

    const int ND  = nNodes * D;
    const int ND4 = ND / 4;

    float* B0 = (float*)d_ws;
    float* B1 = B0 + ND;
    float* B2 = B1 + ND;
    float* bufs[3] = {B0, B1, B2};

    const int nTiles     = (nNodes + 15) / 16;
    const int gemmBlocks = (nTiles + 1) / 2;
    const int propThreads = nEdges * 16;
    const int propBlocks  = (propThreads + 255) / 256;
    const int ndBlocks    = (ND + 255) / 256;
    const int nd4Blocks   = (ND4 + 255) / 256;

    // out = bias ; out += T0 @ W0  (T0 = x)
    cheb_init_bias<<<ndBlocks, 256, 0, stream>>>(out, bias, ND);
    cheb_gemm_acc<<<gemmBlocks, 256, 0, stream>>>(x, W, out, nNodes);

    // T1 = A x ; out += T1 @ W1
    cheb_zero4<<<nd4Blocks, 256, 0, stream>>>((float4*)B0, ND4);
    cheb_propagate<<<propBlocks, 256, 0, stream>>>(x, ei, ew, B0, 1.0f, nEdges);
    if (K > 1)
        cheb_gemm_acc<<<gemmBlocks, 256, 0, stream>>>(B0, W + D * D, out, nNodes);

    // T_k = 2 A T_{k-1} - T_{k-2} ; out += T_k @ W_k
    const float* Tprev = x;
    const float* Tcur  = B0;
    for (int k = 2; k < K; ++k) {
        float* Tnext = bufs[(k - 1) % 3];
        cheb_negcopy4<<<nd4Blocks, 256, 0, stream>>>((float4*)Tnext,
                                                     (const float4*)Tprev, ND4);
        cheb_propagate<<<propBlocks, 256, 0, stream>>>(Tcur, ei, ew, Tnext,
                                                       2.0f, nEdges);
        cheb_gemm_acc<<<gemmBlocks, 256, 0, stream>>>(Tnext, W + (size_t)k * D * D,
                                                      out, nNodes);
        Tprev = Tcur;
        Tcur  = Tnext;
    }
}